// GCN_82291573391755
// MI455X (gfx1250) — compile-verified
//
#include <hip/hip_runtime.h>
#include <math.h>

typedef __attribute__((ext_vector_type(2))) float v2f;
typedef __attribute__((ext_vector_type(8))) float v8f;

#define N_NODES 100000   // divisible by 16 -> WMMA tiles need no row guards
#define N_EDGES 1600000
#define EPS_SM 1e-16f

// ---------------------------------------------------------------------------
// Fused 4-way projection: Yi[N,DOUT] = X[N,DIN] @ Wi[DIN,DOUT] + bi
// (q / k / v / skip share the A operand). One wave32 per 16x16 output tile
// position; K advanced 4/step via V_WMMA_F32_16X16X4_F32, 4 WMMAs per step
// on independent accumulator chains.
//
// VGPR layouts (ISA 7.12.2, 32-bit):
//   A 16x4 : lanes 0-15 rows M=0..15 hold K={k0,k0+1} in {v0,v1};
//            lanes 16-31 same rows hold K={k0+2,k0+3}.
//   B 4x16 : v0 = rows K={k0 (lanes0-15), k0+2 (lanes16-31)}, cols N=lane&15;
//            v1 = rows K={k0+1, k0+3}.
//   C/D    : vgpr i -> row M = i + 8*(lane>=16), col N = lane&15.
// ---------------------------------------------------------------------------
template <int DIN, int DOUT>
__global__ void proj4_wmma(const float* __restrict__ X,
                           const float* __restrict__ W0, const float* __restrict__ b0,
                           float* __restrict__ Y0,
                           const float* __restrict__ W1, const float* __restrict__ b1,
                           float* __restrict__ Y1,
                           const float* __restrict__ W2, const float* __restrict__ b2,
                           float* __restrict__ Y2,
                           const float* __restrict__ W3, const float* __restrict__ b3,
                           float* __restrict__ Y3) {
  const int lane = threadIdx.x;   // 0..31 (wave32)
  const int half = lane >> 4;     // 0 or 1
  const int l    = lane & 15;
  const int row0 = blockIdx.x * 16;
  const int n0   = blockIdx.y * 16;

  const float* xrow = X + (size_t)(row0 + l) * DIN;  // 128B-aligned row base

  v8f acc0 = {}, acc1 = {}, acc2 = {}, acc3 = {};
#pragma unroll
  for (int k0 = 0; k0 < DIN; k0 += 4) {
    const int ka = k0 + 2 * half;                    // even -> 8B aligned
    const v2f a = *(const v2f*)(xrow + ka);          // single b64 load
    const size_t wi0 = (size_t)ka * DOUT + n0 + l;   // row ka
    const size_t wi1 = wi0 + DOUT;                   // row ka+1
    v2f bm;
    bm.x = W0[wi0]; bm.y = W0[wi1];
    acc0 = __builtin_amdgcn_wmma_f32_16x16x4_f32(false, a, false, bm,
                                                 (short)0, acc0, false, false);
    bm.x = W1[wi0]; bm.y = W1[wi1];
    acc1 = __builtin_amdgcn_wmma_f32_16x16x4_f32(false, a, false, bm,
                                                 (short)0, acc1, false, false);
    bm.x = W2[wi0]; bm.y = W2[wi1];
    acc2 = __builtin_amdgcn_wmma_f32_16x16x4_f32(false, a, false, bm,
                                                 (short)0, acc2, false, false);
    bm.x = W3[wi0]; bm.y = W3[wi1];
    acc3 = __builtin_amdgcn_wmma_f32_16x16x4_f32(false, a, false, bm,
                                                 (short)0, acc3, false, false);
  }

  const int   srow = row0 + 8 * half;                // first of this lane's 8 rows
  const int   col  = n0 + l;
  const float bb0 = b0[col], bb1 = b1[col], bb2 = b2[col], bb3 = b3[col];
#pragma unroll
  for (int i = 0; i < 8; ++i) {
    const size_t o = (size_t)(srow + i) * DOUT + col;
    Y0[o] = acc0[i] + bb0;
    Y1[o] = acc1[i] + bb1;
    Y2[o] = acc2[i] + bb2;
    Y3[o] = acc3[i] + bb3;
  }
}

// ---------------------------------------------------------------------------
// Edge phase
// ---------------------------------------------------------------------------
__global__ void init_seg(float* __restrict__ segmax, float* __restrict__ segsum,
                         int n) {
  int i = blockIdx.x * blockDim.x + threadIdx.x;
  if (i < n) {
    segmax[i] = -3.402823466e38f;
    segsum[i] = 0.f;
  }
}

__device__ __forceinline__ void atomicMaxF(float* addr, float val) {
  // monotone bit trick: int-max for >=0, uint-min for <0 (init = -FLT_MAX)
  if (val >= 0.f)
    atomicMax((int*)addr, __float_as_int(val));
  else
    atomicMin((unsigned int*)addr, __float_as_uint(val));
}

template <int D>
__global__ void edge_alpha(const int* __restrict__ src,
                           const int* __restrict__ dst,
                           const float* __restrict__ q,
                           const float* __restrict__ k,
                           float* __restrict__ alpha,
                           float* __restrict__ segmax) {
  int e = blockIdx.x * blockDim.x + threadIdx.x;
  if (e >= N_EDGES) return;
  const int s = src[e], d = dst[e];
  const float4* qp = (const float4*)(q + (size_t)d * D);
  const float4* kp = (const float4*)(k + (size_t)s * D);
  float acc = 0.f;
#pragma unroll
  for (int j = 0; j < D / 4; ++j) {
    float4 a = qp[j], bb = kp[j];
    acc += a.x * bb.x + a.y * bb.y + a.z * bb.z + a.w * bb.w;
  }
  const float al = acc * rsqrtf((float)D);
  alpha[e] = al;
  atomicMaxF(&segmax[d], al);
}

__global__ void edge_exp(const int* __restrict__ dst,
                         float* __restrict__ alpha,
                         const float* __restrict__ segmax,
                         float* __restrict__ segsum) {
  int e = blockIdx.x * blockDim.x + threadIdx.x;
  if (e >= N_EDGES) return;
  const int d = dst[e];
  const float w = __expf(alpha[e] - segmax[d]);
  alpha[e] = w;
  atomicAdd(&segsum[d], w);
}

// D lanes cooperate on one edge: lane j handles feature j -> coalesced v reads
// and coalesced atomic targets within each edge group.
template <int D>
__global__ void edge_agg(const int* __restrict__ src,
                         const int* __restrict__ dst,
                         const float* __restrict__ alpha,
                         const float* __restrict__ segsum,
                         const float* __restrict__ v,
                         float* __restrict__ h) {
  const int tid = blockIdx.x * blockDim.x + threadIdx.x;
  const int e = tid / D;
  const int j = tid % D;
  if (e >= N_EDGES) return;
  const int s = src[e], d = dst[e];
  const float w = alpha[e] / (segsum[d] + EPS_SM);
  atomicAdd(&h[(size_t)d * D + j], w * v[(size_t)s * D + j]);
}

__global__ void relu_k(float* __restrict__ x, int n) {
  int i = blockIdx.x * blockDim.x + threadIdx.x;
  if (i < n) x[i] = fmaxf(x[i], 0.f);
}

__global__ void out_linear(const float* __restrict__ h,
                           const float* __restrict__ Wo,
                           const float* __restrict__ bo,
                           float* __restrict__ out, int n) {
  int i = blockIdx.x * blockDim.x + threadIdx.x;
  if (i >= n) return;
  float a0 = bo[0], a1 = bo[1];
#pragma unroll
  for (int j = 0; j < 16; ++j) {
    const float hv = h[(size_t)i * 16 + j];
    a0 += hv * Wo[j * 2 + 0];
    a1 += hv * Wo[j * 2 + 1];
  }
  out[(size_t)i * 2 + 0] = a0;
  out[(size_t)i * 2 + 1] = a1;
}

// ---------------------------------------------------------------------------
// Orchestration
// ---------------------------------------------------------------------------
extern "C" void kernel_launch(void* const* d_in, const int* in_sizes, int n_in,
                              void* d_out, int out_size, void* d_ws,
                              size_t ws_size, hipStream_t stream) {
  const int*   eidx = (const int*)d_in[0];
  const int*   src  = eidx;              // edge_index[0]
  const int*   dst  = eidx + N_EDGES;    // edge_index[1]
  const float* emb  = (const float*)d_in[1];
  const float* Wq1 = (const float*)d_in[2],  *bq1 = (const float*)d_in[3];
  const float* Wk1 = (const float*)d_in[4],  *bk1 = (const float*)d_in[5];
  const float* Wv1 = (const float*)d_in[6],  *bv1 = (const float*)d_in[7];
  const float* Ws1 = (const float*)d_in[8],  *bs1 = (const float*)d_in[9];
  const float* Wq2 = (const float*)d_in[10], *bq2 = (const float*)d_in[11];
  const float* Wk2 = (const float*)d_in[12], *bk2 = (const float*)d_in[13];
  const float* Wv2 = (const float*)d_in[14], *bv2 = (const float*)d_in[15];
  const float* Ws2 = (const float*)d_in[16], *bs2 = (const float*)d_in[17];
  const float* Wo  = (const float*)d_in[18], *bo  = (const float*)d_in[19];
  float* out = (float*)d_out;

  // workspace layout (floats)
  float* ws = (float*)d_ws;
  const size_t N = N_NODES, E = N_EDGES;
  float* q      = ws;                 // N*32
  float* kk     = q + N * 32;         // N*32
  float* v      = kk + N * 32;        // N*32
  float* h1     = v + N * 32;         // N*32 (skip + agg + relu)
  float* h2     = h1 + N * 32;        // N*16
  float* alpha  = h2 + N * 16;        // E
  float* segmax = alpha + E;          // N
  float* segsum = segmax + N;         // N

  const int TB = 256;
  const int nodeTiles = N_NODES / 16;                         // 6250 exact
  const int eBlk = (N_EDGES + TB - 1) / TB;
  const int nBlk = (N_NODES + TB - 1) / TB;

  // ---------------- layer 1: TransformerConv(32 -> 32) ----------------
  proj4_wmma<32, 32><<<dim3(nodeTiles, 2), 32, 0, stream>>>(
      emb, Wq1, bq1, q, Wk1, bk1, kk, Wv1, bv1, v, Ws1, bs1, h1);
  init_seg<<<nBlk, TB, 0, stream>>>(segmax, segsum, N_NODES);
  edge_alpha<32><<<eBlk, TB, 0, stream>>>(src, dst, q, kk, alpha, segmax);
  edge_exp<<<eBlk, TB, 0, stream>>>(dst, alpha, segmax, segsum);
  edge_agg<32><<<(N_EDGES * 32 + TB - 1) / TB, TB, 0, stream>>>(
      src, dst, alpha, segsum, v, h1);
  relu_k<<<(N_NODES * 32 + TB - 1) / TB, TB, 0, stream>>>(h1, N_NODES * 32);

  // ---------------- layer 2: TransformerConv(32 -> 16) ----------------
  proj4_wmma<32, 16><<<dim3(nodeTiles, 1), 32, 0, stream>>>(
      h1, Wq2, bq2, q, Wk2, bk2, kk, Wv2, bv2, v, Ws2, bs2, h2);
  init_seg<<<nBlk, TB, 0, stream>>>(segmax, segsum, N_NODES);
  edge_alpha<16><<<eBlk, TB, 0, stream>>>(src, dst, q, kk, alpha, segmax);
  edge_exp<<<eBlk, TB, 0, stream>>>(dst, alpha, segmax, segsum);
  edge_agg<16><<<(N_EDGES * 16 + TB - 1) / TB, TB, 0, stream>>>(
      src, dst, alpha, segsum, v, h2);
  relu_k<<<(N_NODES * 16 + TB - 1) / TB, TB, 0, stream>>>(h2, N_NODES * 16);

  // ---------------- output linear 16 -> 2 ----------------
  out_linear<<<nBlk, TB, 0, stream>>>(h2, Wo, bo, out, N_NODES);
}